// DifferentiableKendallTau_81123342287264
// MI455X (gfx1250) — compile-verified
//
#include <hip/hip_runtime.h>

typedef __attribute__((ext_vector_type(2))) float v2f;
typedef __attribute__((ext_vector_type(8))) float v8f;

#define N_ELEM 8192
#define N_TILE 512                    // 8192 / 16 tiles per dimension
#define WAVES_TOTAL 8192              // 1024 blocks * 8 waves
#define TILES_PER_WAVE 32             // 512*512 / 8192

// gfx1250 hardware tanh (TRANS pipe, co-executes with VALU).
__device__ __forceinline__ float hw_tanh(float x) {
#if __has_builtin(__builtin_amdgcn_tanhf)
    return __builtin_amdgcn_tanhf(x);
#else
    float y;
    asm("v_tanh_f32 %0, %1" : "=v"(y) : "v"(x));
    return y;
#endif
}

// X[i,j] = (Pj-Pi)(Tj-Ti) with P = 10*p (temperature folded in) is rank-4:
//   X = C*1^T + 1*C^T - P*T^T - T*P^T ,  C = P .* T
// Each 16x16 tile of X is one V_WMMA_F32_16X16X4_F32 with
//   A row i = [Ci, 1, Pi, Ti],  B col j = [1, Cj, -Tj, -Pj].
// Each wave fixes one column tile J (B hoisted out of the loop) and strides
// over 32 row tiles I; tanh is applied to the 8 accumulator VGPRs per tile.
__global__ void __launch_bounds__(256)
kendall_tau_wmma(const float* __restrict__ pred,
                 const float* __restrict__ target,
                 float* __restrict__ partial) {
    __shared__ float sP[N_ELEM];  // 32 KB
    __shared__ float sT[N_ELEM];  // 32 KB (contiguous after sP -> 2addr loads)

    for (int idx = threadIdx.x; idx < N_ELEM; idx += blockDim.x) {
        sP[idx] = pred[idx] * 10.0f;   // fold 1/TEMPERATURE = 10 into P
        sT[idx] = target[idx];
    }
    __syncthreads();

    const int lane = threadIdx.x & 31;
    // Wave-uniform ID in an SGPR so all loop control is scalar and EXEC is
    // statically all-1s at every WMMA (ISA 7.12 requirement).
    const int waveId = __builtin_amdgcn_readfirstlane(
        (int)(blockIdx.x * (blockDim.x >> 5)) + (threadIdx.x >> 5));

    const int J      = waveId & (N_TILE - 1);  // fixed column tile
    const int Igroup = waveId >> 9;            // 0..15; I = Igroup + 16*k

    // A-matrix 16x4 f32 layout (ISA 7.12.2): lanes 0-15 carry K={0,1},
    // lanes 16-31 carry K={2,3}, row M = lane & 15. B mirrors with N = lane&15.
    const int half = lane >> 4;
    const int mn   = lane & 15;

    // Hoisted B operand for column tile J.
    const int   j  = (J << 4) + mn;
    const float Pj = sP[j], Tj = sT[j];
    v2f b;
    b.x = half ? -Tj : 1.0f;
    b.y = half ? -Pj : Pj * Tj;

    float acc = 0.0f;
    const int ibase = (Igroup << 4) + mn;      // i strides by 256 per step

#pragma unroll 4
    for (int k = 0; k < TILES_PER_WAVE; ++k) {
        const int   i  = ibase + (k << 8);
        const float Pi = sP[i], Ti = sT[i];    // one ds_load_2addr pair

        v2f a;
        a.x = half ? Pi : Pi * Ti;
        a.y = half ? Ti : 1.0f;

        v8f c = {};
        c = __builtin_amdgcn_wmma_f32_16x16x4_f32(
                /*neg_a=*/false, a, /*neg_b=*/false, b,
                /*c_mod=*/(short)0, c, /*reuse_a=*/false, /*reuse_b=*/false);

#pragma unroll
        for (int r = 0; r < 8; ++r)
            acc += hw_tanh(c[r]);              // diagonal gives tanh(0)=0
    }

    // wave32 reduction, then one global atomic per wave
#pragma unroll
    for (int off = 16; off > 0; off >>= 1)
        acc += __shfl_down(acc, off, 32);
    if (lane == 0)
        atomicAdd(partial, acc);
}

__global__ void kendall_tau_finalize(const float* __restrict__ partial,
                                     float* __restrict__ out) {
    if (threadIdx.x == 0 && blockIdx.x == 0) {
        const float inv_pairs = 1.0f / ((float)N_ELEM * (float)(N_ELEM - 1));
        out[0] = partial[0] * inv_pairs;   // full-matrix sum / n(n-1)
    }
}

extern "C" void kernel_launch(void* const* d_in, const int* in_sizes, int n_in,
                              void* d_out, int out_size, void* d_ws, size_t ws_size,
                              hipStream_t stream) {
    (void)in_sizes; (void)n_in; (void)out_size; (void)ws_size;
    const float* pred   = (const float*)d_in[0];
    const float* target = (const float*)d_in[1];
    float* out     = (float*)d_out;
    float* partial = (float*)d_ws;

    // Zero the accumulator every call (graph-capture-safe async memset; the
    // harness does not re-poison between replays).
    hipMemsetAsync(partial, 0, sizeof(float), stream);

    dim3 grid(WAVES_TOTAL / 8), block(256);   // 8192 waves, 32 tiles each
    kendall_tau_wmma<<<grid, block, 0, stream>>>(pred, target, partial);
    kendall_tau_finalize<<<1, 64, 0, stream>>>(partial, out);
}